// SimpleRelativeLayer_72035191488656
// MI455X (gfx1250) — compile-verified
//
#include <hip/hip_runtime.h>

#define RADIUS_F 0.22f
#define R2_F (RADIUS_F * RADIUS_F)

typedef __attribute__((ext_vector_type(2))) float v2f;
typedef __attribute__((ext_vector_type(8))) float v8f;
typedef __attribute__((ext_vector_type(4))) unsigned u32x4;
typedef __attribute__((ext_vector_type(4))) int i32x4;
typedef __attribute__((ext_vector_type(8))) int i32x8;

// D = A(16x4) * B(4x16) + C(16x16), fp32, CDNA5 WMMA.
__device__ __forceinline__ v8f wmma4(v2f a, v2f b, v8f c) {
  return __builtin_amdgcn_wmma_f32_16x16x4_f32(false, a, false, b, (short)0, c,
                                               false, false);
}

// ---------------------------------------------------------------------------
// Tensor Data Mover: 1-D fp32 tile (whole weight matrix) global -> LDS.
// D# layout per CDNA5 ISA ch.8: group0 {count=1, lds_addr, global_addr,
// type=2}; group1 {data_size=4B, tensor_dim0=tile_dim0=nelem, tensor_dim1=1,
// stride=nelem}. Tracked by TENSORcnt.
// ---------------------------------------------------------------------------
__device__ __forceinline__ void tdm_load_f32(const float* gptr,
                                             const float* lds_ptr,
                                             unsigned nelem) {
  unsigned long long ga = (unsigned long long)(uintptr_t)gptr;
  unsigned lds_off = (unsigned)(unsigned long long)(uintptr_t)lds_ptr;

  u32x4 g0;
  g0[0] = 1u;                                   // count=1 (valid user D#)
  g0[1] = lds_off;                              // lds_addr (bytes)
  g0[2] = (unsigned)(ga & 0xffffffffull);       // global_addr[31:0]
  g0[3] = (unsigned)((ga >> 32) & 0x1ffffffull) // global_addr[56:32]
          | (2u << 30);                         // type=2 ("image")

  unsigned long long nel = nelem;
  unsigned long long q0 = (2ull << 16)                       // data_size=4B
                          | ((nel & 0xffffull) << 48);       // tensor_dim0 lo16
  unsigned long long q1 = ((nel >> 16) & 0xffffull)          // tensor_dim0 hi16
                          | (1ull << 16)                     // tensor_dim1 = 1
                          | ((nel & 0xffffull) << 48);       // tile_dim0
  unsigned long long q2 = nel << 32;  // tensor_dim0_stride lo32 (bits 191:160)
  unsigned long long q3 = 0;          // stride hi16 = 0 (nelem < 2^32)

  i32x8 g1;
  g1[0] = (int)q0; g1[1] = (int)(q0 >> 32);
  g1[2] = (int)q1; g1[3] = (int)(q1 >> 32);
  g1[4] = (int)q2; g1[5] = (int)(q2 >> 32);
  g1[6] = (int)q3; g1[7] = (int)(q3 >> 32);

  i32x4 z4 = {0, 0, 0, 0};
#if __clang_major__ >= 23
  i32x8 z8 = {0, 0, 0, 0, 0, 0, 0, 0};
  __builtin_amdgcn_tensor_load_to_lds(g0, g1, z4, z4, z8, 0);
#else
  __builtin_amdgcn_tensor_load_to_lds(g0, g1, z4, z4, 0);
#endif
}

// ---------------------------------------------------------------------------
// Kernel 1: farthest point sampling. Single persistent workgroup; min-dist
// array lives in global scratch (L2-resident), argmax via LDS tree reduce.
// ---------------------------------------------------------------------------
__global__ __launch_bounds__(1024) void fps_kernel(
    const float* __restrict__ points, int N, int S,
    int* __restrict__ sample_inds, float* __restrict__ samples,
    float* __restrict__ mind) {
  __shared__ float sv[1024];
  __shared__ int si[1024];
  const int t = threadIdx.x;

  for (int i = t; i < N; i += 1024) mind[i] = 3.4e38f;
  if (t == 0) {
    sample_inds[0] = 0;
    samples[0] = points[0];
    samples[1] = points[1];
    samples[2] = points[2];
  }
  __syncthreads();

  for (int s = 1; s < S; s++) {
    const float px = samples[3 * (s - 1) + 0];
    const float py = samples[3 * (s - 1) + 1];
    const float pz = samples[3 * (s - 1) + 2];
    float bv = -1.0f;
    int bi = 0;
    for (int i = t; i < N; i += 1024) {
      __builtin_prefetch(&points[3 * (i + 1024)], 0, 0);
      float dx = points[3 * i + 0] - px;
      float dy = points[3 * i + 1] - py;
      float dz = points[3 * i + 2] - pz;
      float d = dx * dx + dy * dy + dz * dz;
      float m = fminf(mind[i], d);
      mind[i] = m;
      if (m > bv) { bv = m; bi = i; }  // strict > keeps first index per thread
    }
    sv[t] = bv;
    si[t] = bi;
    __syncthreads();
    for (int off = 512; off > 0; off >>= 1) {
      if (t < off) {
        float ov = sv[t + off];
        int oi = si[t + off];
        if (ov > sv[t] || (ov == sv[t] && oi < si[t])) { sv[t] = ov; si[t] = oi; }
      }
      __syncthreads();
    }
    if (t == 0) {
      int w = si[0];
      sample_inds[s] = w;
      samples[3 * s + 0] = points[3 * w + 0];
      samples[3 * s + 1] = points[3 * w + 1];
      samples[3 * s + 2] = points[3 * w + 2];
    }
    __syncthreads();
  }
}

// ---------------------------------------------------------------------------
// Kernel 2: radius-limited kNN (k=nb) by iterative min-extraction.
// ---------------------------------------------------------------------------
__global__ __launch_bounds__(256) void ball_kernel(
    const float* __restrict__ points, int N, int nb,
    const float* __restrict__ samples, int* __restrict__ rad_inds) {
  __shared__ int sel[32];
  __shared__ float sv[256];
  __shared__ int si[256];
  const int s = blockIdx.x, t = threadIdx.x;
  const float px = samples[3 * s + 0];
  const float py = samples[3 * s + 1];
  const float pz = samples[3 * s + 2];

  for (int r = 0; r < nb; r++) {
    float bv = 3.4e38f;
    int bi = -1;
    for (int i = t; i < N; i += 256) {
      float dx = points[3 * i + 0] - px;
      float dy = points[3 * i + 1] - py;
      float dz = points[3 * i + 2] - pz;
      float d = dx * dx + dy * dy + dz * dz;
      if (d <= R2_F) {
        bool used = false;
        for (int q = 0; q < r; q++) used = used || (sel[q] == i);
        if (!used && d < bv) { bv = d; bi = i; }
      }
    }
    sv[t] = bv;
    si[t] = bi;
    __syncthreads();
    for (int off = 128; off > 0; off >>= 1) {
      if (t < off) {
        float ov = sv[t + off];
        int oi = si[t + off];
        if (oi >= 0 && (si[t] < 0 || ov < sv[t] || (ov == sv[t] && oi < si[t]))) {
          sv[t] = ov;
          si[t] = oi;
        }
      }
      __syncthreads();
    }
    if (t == 0) {
      int w = (si[0] >= 0) ? si[0] : sel[0];
      sel[r] = w;
      rad_inds[s * nb + r] = w;
    }
    __syncthreads();
  }
}

// ---------------------------------------------------------------------------
// Kernel 3: emit rad_points + rad_cluster portions of the output.
// ---------------------------------------------------------------------------
__global__ void gather_out_kernel(const float* __restrict__ points,
                                  const int* __restrict__ rad_inds,
                                  float* __restrict__ out, int total, int nb) {
  int i = blockIdx.x * blockDim.x + threadIdx.x;
  if (i < total) {
    int idx = rad_inds[i];
    out[3 * i + 0] = points[3 * idx + 0];
    out[3 * i + 1] = points[3 * idx + 1];
    out[3 * i + 2] = points[3 * idx + 2];
    out[3 * total + i] = (float)(i / nb);
  }
}

__global__ void zero_kernel(int* __restrict__ p, int n) {
  int i = blockIdx.x * blockDim.x + threadIdx.x;
  if (i < n) p[i] = 0;
}

// ---------------------------------------------------------------------------
// Kernel 4: encoder MLP 3->80->40->20 via V_WMMA_F32_16X16X4_F32.
// Weights/biases staged global->LDS once per block by the Tensor Data Mover;
// per-cluster max-pool fused via float-bit atomicMax (values >= 0 post-relu).
// ---------------------------------------------------------------------------
#define EW1 0
#define EW2 240
#define EW3 3440
#define EB1 4240
#define EB2 4320
#define EB3 4360
#define EWTOT 4380

__global__ __launch_bounds__(128) void encoder_kernel(
    const float* __restrict__ points, const float* __restrict__ samples,
    const int* __restrict__ rad_inds, const float* __restrict__ w1,
    const float* __restrict__ b1, const float* __restrict__ w2,
    const float* __restrict__ b2, const float* __restrict__ w3,
    const float* __restrict__ b3, int* __restrict__ featBits, int totalRows,
    int nb) {
  __shared__ float h1[4][16][80];
  __shared__ float h2[4][16][40];
  __shared__ float wsh[EWTOT];
  const int t = threadIdx.x;
  const int w = t >> 5, lane = t & 31, half = lane >> 4, nl = lane & 15;
  const int row0 = (blockIdx.x * 4 + w) * 16;

  if (w == 0) {  // wave 0 drives the TDM (EXEC-independent, per-wave issue)
    tdm_load_f32(w1, &wsh[EW1], 240);
    tdm_load_f32(w2, &wsh[EW2], 3200);
    tdm_load_f32(w3, &wsh[EW3], 800);
    tdm_load_f32(b1, &wsh[EB1], 80);
    tdm_load_f32(b2, &wsh[EB2], 40);
    tdm_load_f32(b3, &wsh[EB3], 20);
    __builtin_amdgcn_s_wait_tensorcnt(0);
  }

  // ---- Layer 1 A-fragment: relative[16x3 pad 4]
  float r0 = 0.f, r1 = 0.f, r2 = 0.f;
  {
    int m = row0 + nl;
    if (m < totalRows) {
      int s = m / nb;
      int idx = rad_inds[m];
      const float inv = 1.0f / RADIUS_F;
      r0 = (points[3 * idx + 0] - samples[3 * s + 0]) * inv;
      r1 = (points[3 * idx + 1] - samples[3 * s + 1]) * inv;
      r2 = (points[3 * idx + 2] - samples[3 * s + 2]) * inv;
    }
  }
  __syncthreads();  // weights visible to all waves

  v2f a1;
  a1.x = half ? r2 : r0;    // A[m][2*half]
  a1.y = half ? 0.0f : r1;  // A[m][2*half+1]  (K=3 padded to 4)
#pragma unroll
  for (int nt = 0; nt < 5; nt++) {
    int col = nt * 16 + nl;
    v2f b;
    b.x = wsh[EW1 + (2 * half) * 80 + col];
    b.y = half ? 0.0f : wsh[EW1 + 80 + col];
    float bias = wsh[EB1 + col];
    v8f c;
#pragma unroll
    for (int j = 0; j < 8; j++) c[j] = bias;
    c = wmma4(a1, b, c);
#pragma unroll
    for (int j = 0; j < 8; j++) h1[w][j + 8 * half][col] = fmaxf(c[j], 0.0f);
  }
  __syncthreads();

  // ---- Layer 2: h1[16x80] x w2[80x40] -> relu -> h2
#pragma unroll
  for (int nt = 0; nt < 3; nt++) {
    int col = nt * 16 + nl;
    bool cv = col < 40;
    int ccol = cv ? col : 0;  // clamp: harmless lanes, results discarded
    float bias = cv ? wsh[EB2 + ccol] : 0.0f;
    v8f c;
#pragma unroll
    for (int j = 0; j < 8; j++) c[j] = bias;
#pragma unroll
    for (int k = 0; k < 20; k++) {
      int kb = 4 * k + 2 * half;
      v2f a;
      a.x = h1[w][nl][kb];
      a.y = h1[w][nl][kb + 1];
      v2f b;
      b.x = cv ? wsh[EW2 + kb * 40 + ccol] : 0.0f;
      b.y = cv ? wsh[EW2 + (kb + 1) * 40 + ccol] : 0.0f;
      c = wmma4(a, b, c);
    }
    if (cv) {
#pragma unroll
      for (int j = 0; j < 8; j++) h2[w][j + 8 * half][col] = fmaxf(c[j], 0.0f);
    }
  }
  __syncthreads();

  // ---- Layer 3: h2[16x40] x w3[40x20] -> relu -> atomic maxpool into feat
#pragma unroll
  for (int nt = 0; nt < 2; nt++) {
    int col = nt * 16 + nl;
    bool cv = col < 20;
    int ccol = cv ? col : 0;
    float bias = cv ? wsh[EB3 + ccol] : 0.0f;
    v8f c;
#pragma unroll
    for (int j = 0; j < 8; j++) c[j] = bias;
#pragma unroll
    for (int k = 0; k < 10; k++) {
      int kb = 4 * k + 2 * half;
      v2f a;
      a.x = h2[w][nl][kb];
      a.y = h2[w][nl][kb + 1];
      v2f b;
      b.x = cv ? wsh[EW3 + kb * 20 + ccol] : 0.0f;
      b.y = cv ? wsh[EW3 + (kb + 1) * 20 + ccol] : 0.0f;
      c = wmma4(a, b, c);
    }
#pragma unroll
    for (int j = 0; j < 8; j++) {
      int m = row0 + j + 8 * half;
      if (cv && m < totalRows) {
        float v = fmaxf(c[j], 0.0f);
        atomicMax(&featBits[(m / nb) * 20 + col], __float_as_int(v));
      }
    }
  }
}

// ---------------------------------------------------------------------------
// Kernel 5: decoder MLP 20->40->80->75 via WMMA; TDM-staged weights;
// epilogue applies *RADIUS + midpoint and writes resized_deco into d_out.
// ---------------------------------------------------------------------------
#define DW1 0
#define DW2 800
#define DW3 4000
#define DB1 10000
#define DB2 10040
#define DB3 10120
#define DWTOT 10195

__global__ __launch_bounds__(128) void decoder_kernel(
    const int* __restrict__ featBits, const float* __restrict__ samples,
    const float* __restrict__ w1, const float* __restrict__ b1,
    const float* __restrict__ w2, const float* __restrict__ b2,
    const float* __restrict__ w3, const float* __restrict__ b3,
    float* __restrict__ out_deco, int S) {
  __shared__ float g1[4][16][40];
  __shared__ float g2[4][16][80];
  __shared__ float wsh[DWTOT];
  const int t = threadIdx.x;
  const int w = t >> 5, lane = t & 31, half = lane >> 4, nl = lane & 15;
  const int row0 = (blockIdx.x * 4 + w) * 16;
  const int mrow = row0 + nl;
  const bool rv = mrow < S;
  const float* feat = (const float*)featBits;

  if (w == 0) {
    tdm_load_f32(w1, &wsh[DW1], 800);
    tdm_load_f32(w2, &wsh[DW2], 3200);
    tdm_load_f32(w3, &wsh[DW3], 6000);
    tdm_load_f32(b1, &wsh[DB1], 40);
    tdm_load_f32(b2, &wsh[DB2], 80);
    tdm_load_f32(b3, &wsh[DB3], 75);
    __builtin_amdgcn_s_wait_tensorcnt(0);
  }
  __syncthreads();

  // ---- L1: feat[16x20] x w1[20x40] -> relu -> g1
#pragma unroll
  for (int nt = 0; nt < 3; nt++) {
    int col = nt * 16 + nl;
    bool cv = col < 40;
    int ccol = cv ? col : 0;
    float bias = cv ? wsh[DB1 + ccol] : 0.0f;
    v8f c;
#pragma unroll
    for (int j = 0; j < 8; j++) c[j] = bias;
#pragma unroll
    for (int k = 0; k < 5; k++) {
      int kb = 4 * k + 2 * half;
      v2f a;
      a.x = rv ? feat[mrow * 20 + kb] : 0.0f;
      a.y = rv ? feat[mrow * 20 + kb + 1] : 0.0f;
      v2f b;
      b.x = cv ? wsh[DW1 + kb * 40 + ccol] : 0.0f;
      b.y = cv ? wsh[DW1 + (kb + 1) * 40 + ccol] : 0.0f;
      c = wmma4(a, b, c);
    }
    if (cv) {
#pragma unroll
      for (int j = 0; j < 8; j++) g1[w][j + 8 * half][col] = fmaxf(c[j], 0.0f);
    }
  }
  __syncthreads();

  // ---- L2: g1[16x40] x w2[40x80] -> relu -> g2
#pragma unroll
  for (int nt = 0; nt < 5; nt++) {
    int col = nt * 16 + nl;
    float bias = wsh[DB2 + col];
    v8f c;
#pragma unroll
    for (int j = 0; j < 8; j++) c[j] = bias;
#pragma unroll
    for (int k = 0; k < 10; k++) {
      int kb = 4 * k + 2 * half;
      v2f a;
      a.x = g1[w][nl][kb];
      a.y = g1[w][nl][kb + 1];
      v2f b;
      b.x = wsh[DW2 + kb * 80 + col];
      b.y = wsh[DW2 + (kb + 1) * 80 + col];
      c = wmma4(a, b, c);
    }
#pragma unroll
    for (int j = 0; j < 8; j++) g2[w][j + 8 * half][col] = fmaxf(c[j], 0.0f);
  }
  __syncthreads();

  // ---- L3: g2[16x80] x w3[80x75] (no relu) -> scale + midpoint -> out
#pragma unroll
  for (int nt = 0; nt < 5; nt++) {
    int col = nt * 16 + nl;
    bool cv = col < 75;
    int ccol = cv ? col : 0;
    float bias = cv ? wsh[DB3 + ccol] : 0.0f;
    v8f c;
#pragma unroll
    for (int j = 0; j < 8; j++) c[j] = bias;
#pragma unroll
    for (int k = 0; k < 20; k++) {
      int kb = 4 * k + 2 * half;
      v2f a;
      a.x = g2[w][nl][kb];
      a.y = g2[w][nl][kb + 1];
      v2f b;
      b.x = cv ? wsh[DW3 + kb * 75 + ccol] : 0.0f;
      b.y = cv ? wsh[DW3 + (kb + 1) * 75 + ccol] : 0.0f;
      c = wmma4(a, b, c);
    }
#pragma unroll
    for (int j = 0; j < 8; j++) {
      int r = row0 + j + 8 * half;
      if (cv && r < S) {
        out_deco[r * 75 + col] = c[j] * RADIUS_F + samples[3 * r + (col % 3)];
      }
    }
  }
}

// ---------------------------------------------------------------------------
extern "C" void kernel_launch(void* const* d_in, const int* in_sizes, int n_in,
                              void* d_out, int out_size, void* d_ws,
                              size_t ws_size, hipStream_t stream) {
  const float* points = (const float*)d_in[0];
  const float* ew1 = (const float*)d_in[1];
  const float* eb1 = (const float*)d_in[2];
  const float* ew2 = (const float*)d_in[3];
  const float* eb2 = (const float*)d_in[4];
  const float* ew3 = (const float*)d_in[5];
  const float* eb3 = (const float*)d_in[6];
  const float* dw1 = (const float*)d_in[7];
  const float* db1 = (const float*)d_in[8];
  const float* dw2 = (const float*)d_in[9];
  const float* db2 = (const float*)d_in[10];
  const float* dw3 = (const float*)d_in[11];
  const float* db3 = (const float*)d_in[12];

  const int N = in_sizes[0] / 3;    // 65536
  const int nb = in_sizes[12] / 3;  // 25
  const int S = (N + nb - 1) / nb;  // 2622
  const int total = S * nb;         // 65550

  auto align16 = [](size_t x) { return (x + 15) & ~(size_t)15; };
  char* ws = (char*)d_ws;
  int* sample_inds = (int*)ws;  ws += align16((size_t)S * 4);
  float* samples = (float*)ws;  ws += align16((size_t)S * 3 * 4);
  int* rad_inds = (int*)ws;     ws += align16((size_t)total * 4);
  int* featBits = (int*)ws;     ws += align16((size_t)S * 20 * 4);
  float* mind = (float*)ws;     ws += align16((size_t)N * 4);

  float* out = (float*)d_out;
  float* out_deco = out + (size_t)total * 4;  // after rad_points + rad_cluster

  fps_kernel<<<1, 1024, 0, stream>>>(points, N, S, sample_inds, samples, mind);
  ball_kernel<<<S, 256, 0, stream>>>(points, N, nb, samples, rad_inds);
  gather_out_kernel<<<(total + 255) / 256, 256, 0, stream>>>(points, rad_inds,
                                                             out, total, nb);
  zero_kernel<<<(S * 20 + 255) / 256, 256, 0, stream>>>(featBits, S * 20);

  const int encTiles = (total + 15) / 16;
  encoder_kernel<<<(encTiles + 3) / 4, 128, 0, stream>>>(
      points, samples, rad_inds, ew1, eb1, ew2, eb2, ew3, eb3, featBits, total,
      nb);

  const int decTiles = (S + 15) / 16;
  decoder_kernel<<<(decTiles + 3) / 4, 128, 0, stream>>>(
      featBits, samples, dw1, db1, dw2, db2, dw3, db3, out_deco, S);
}